// Cluster_GT_15169824489966
// MI455X (gfx1250) — compile-verified
//
#include <hip/hip_runtime.h>
#include <hip/hip_bf16.h>

typedef __attribute__((ext_vector_type(2))) float v2f;
typedef __attribute__((ext_vector_type(8))) float v8f;

#define DD 64
#define NNODES 65536
#define NEDGE 16384
#define NPATCH_TOT 1024
#define NBATCH 32
#define MAXE 2048

// ---------------------------------------------------------------------------
// WMMA helpers: V_WMMA_F32_16X16X4_F32, fp32 in/out (faithful to reference).
// A (16x4 f32): lane = kh*16+m holds A[m][2kh], A[m][2kh+1]
// B (4x16 f32): vgpr r: lanes0-15 -> K=r, lanes16-31 -> K=r+2 ; N = lane&15
// C/D (16x16 f32): vgpr r: lanes0-15 -> M=r, lanes16-31 -> M=8+r ; N = lane&15
// ---------------------------------------------------------------------------
__device__ __forceinline__ v8f wmma4(v2f a, v2f b, v8f c) {
  return __builtin_amdgcn_wmma_f32_16x16x4_f32(false, a, false, b, (short)0, c,
                                               false, false);
}

__device__ __forceinline__ void zero_acc4(v8f acc[4]) {
#pragma unroll
  for (int t = 0; t < 4; ++t)
#pragma unroll
    for (int r = 0; r < 8; ++r) acc[t][r] = 0.0f;
}

// 16x64 row-tile times 64x64 weight (row-major W[k][n]), K=64.
// Arow = per-lane pointer to row (row0 + (lane&15)) of A, contiguous 64 floats.
__device__ __forceinline__ void gemm_rowtile(const float* __restrict__ Arow,
                                             const float* __restrict__ W,
                                             v8f acc[4]) {
  const int lane = threadIdx.x & 31;
  const int kh = lane >> 4;      // 0 or 1
  const int n0 = lane & 15;
#pragma unroll
  for (int kb = 0; kb < 16; ++kb) {
    const int ka = kb * 4 + 2 * kh;
    v2f a;
    a.x = Arow[ka];
    a.y = Arow[ka + 1];
#pragma unroll
    for (int t = 0; t < 4; ++t) {
      v2f b;
      const float* Wc = W + t * 16 + n0;
      b.x = Wc[(size_t)ka * DD];
      b.y = Wc[(size_t)(ka + 1) * DD];
      acc[t] = wmma4(a, b, acc[t]);
    }
  }
}

// Store 16x64 accumulator tile (4 x 16x16) to dst (row stride ld), +bias, relu.
__device__ __forceinline__ void store_acc(float* __restrict__ dst, int ld,
                                          const v8f acc[4],
                                          const float* __restrict__ bias,
                                          bool doRelu) {
  const int lane = threadIdx.x & 31;
  const int n0 = lane & 15;
  const int mhi = (lane >> 4) * 8;
#pragma unroll
  for (int t = 0; t < 4; ++t) {
#pragma unroll
    for (int r = 0; r < 8; ++r) {
      const int nn = t * 16 + n0;
      float v = acc[t][r] + bias[nn];
      if (doRelu) v = fmaxf(v, 0.0f);
      dst[(size_t)(mhi + r) * ld + nn] = v;
    }
  }
}

// ---------------------------------------------------------------------------
// Small setup kernels
// ---------------------------------------------------------------------------
__global__ void zero_kernel(float* p, int n) {
  int i = blockIdx.x * blockDim.x + threadIdx.x;
  if (i < n) p[i] = 0.0f;
}

__global__ void deg_accum_kernel(const float* __restrict__ attr,
                                 const int* __restrict__ rown,
                                 float* __restrict__ deg) {
  int e = blockIdx.x * blockDim.x + threadIdx.x;
  if (e < NEDGE) atomicAdd(&deg[rown[e]], attr[e]);
}

__global__ void attr_norm_kernel(const float* __restrict__ attr,
                                 const int* __restrict__ rown,
                                 const int* __restrict__ coln,
                                 const float* __restrict__ deg,
                                 float* __restrict__ attrn) {
  int e = blockIdx.x * blockDim.x + threadIdx.x;
  if (e >= NEDGE) return;
  float dr = deg[rown[e]], dc = deg[coln[e]];
  float ir = dr > 0.0f ? rsqrtf(fmaxf(dr, 1e-6f)) : 0.0f;
  float ic = dc > 0.0f ? rsqrtf(fmaxf(dc, 1e-6f)) : 0.0f;
  attrn[e] = ir * attr[e] * ic;
}

// start[p] = first index i with batch[i] >= p (batch is sorted). p in [0,P].
__global__ void patch_start_kernel(const int* __restrict__ batch,
                                   int* __restrict__ startp) {
  int p = blockIdx.x * blockDim.x + threadIdx.x;
  if (p > NPATCH_TOT) return;
  int lo = 0, hi = NNODES;
  while (lo < hi) {
    int mid = (lo + hi) >> 1;
    if (batch[mid] < p) lo = mid + 1; else hi = mid;
  }
  startp[p] = lo;
}

// ---------------------------------------------------------------------------
// Fused 2-layer MLP with ReLU after both layers: out = relu(relu(in@W1+b1)@W2+b2)
// 128 threads = 4 waves, one 16-row tile per wave; LDS round-trip between GEMMs.
// Used for input_transform (rows=65536) and post-deepset+relu (rows=1024).
// ---------------------------------------------------------------------------
__global__ void mlp2rr_kernel(const float* __restrict__ in,
                              const float* __restrict__ W1,
                              const float* __restrict__ b1,
                              const float* __restrict__ W2,
                              const float* __restrict__ b2,
                              float* __restrict__ out) {
  __shared__ float lds[4 * 16 * DD];
  const int wave = threadIdx.x >> 5;
  const int lane = threadIdx.x & 31;
  const int row0 = (blockIdx.x * 4 + wave) * 16;
  const int m = lane & 15;
  float* tl = lds + wave * (16 * DD);

  v8f acc[4];
  zero_acc4(acc);
  gemm_rowtile(in + (size_t)(row0 + m) * DD, W1, acc);
  store_acc(tl, DD, acc, b1, true);
  __syncthreads();
  zero_acc4(acc);
  gemm_rowtile(tl + m * DD, W2, acc);
  store_acc(out + (size_t)row0 * DD, DD, acc, b2, true);
}

// ---------------------------------------------------------------------------
// Gather hs = h[mapper] and project: Q = relu(hs@Wpre1+b)@Wpre2+b (no final relu)
// K = hs@Wk+bk ; V = hs@Wv+bv
// ---------------------------------------------------------------------------
__global__ void qkv_kernel(const float* __restrict__ h,
                           const int* __restrict__ mapper,
                           const float* __restrict__ Wpre1, const float* __restrict__ bpre1,
                           const float* __restrict__ Wpre2, const float* __restrict__ bpre2,
                           const float* __restrict__ Wk, const float* __restrict__ bk,
                           const float* __restrict__ Wv, const float* __restrict__ bv,
                           float* __restrict__ Qn, float* __restrict__ Kn,
                           float* __restrict__ Vn) {
  __shared__ float lds[4 * 16 * DD];
  const int wave = threadIdx.x >> 5;
  const int lane = threadIdx.x & 31;
  const int row0 = (blockIdx.x * 4 + wave) * 16;
  const int m = lane & 15;
  const float* Arow = h + (size_t)mapper[row0 + m] * DD;
  float* tl = lds + wave * (16 * DD);

  v8f acc[4];
  // Q path (chained)
  zero_acc4(acc);
  gemm_rowtile(Arow, Wpre1, acc);
  store_acc(tl, DD, acc, bpre1, true);
  __syncthreads();
  zero_acc4(acc);
  gemm_rowtile(tl + m * DD, Wpre2, acc);
  store_acc(Qn + (size_t)row0 * DD, DD, acc, bpre2, false);
  // K
  zero_acc4(acc);
  gemm_rowtile(Arow, Wk, acc);
  store_acc(Kn + (size_t)row0 * DD, DD, acc, bk, false);
  // V
  zero_acc4(acc);
  gemm_rowtile(Arow, Wv, acc);
  store_acc(Vn + (size_t)row0 * DD, DD, acc, bv, false);
}

// ---------------------------------------------------------------------------
// Per-patch reduction: one wave per patch.
//   Qs[p]  = sum Q rows           (raw, post-deepset MLP done later)
//   Ksk[p] = relu(mean K rows)
//   M[p]   = relu(K)^T @ V   (64x64, K-dim = nodes of patch) via WMMA
// ---------------------------------------------------------------------------
__global__ void patch_reduce_kernel(const float* __restrict__ Qn,
                                    const float* __restrict__ Kn,
                                    const float* __restrict__ Vn,
                                    const int* __restrict__ startp,
                                    float* __restrict__ M,
                                    float* __restrict__ Qs,
                                    float* __restrict__ Ksk) {
  const int p = blockIdx.x;
  const int s = startp[p], e = startp[p + 1];
  const int lane = threadIdx.x;

  // feature sums (lane owns features lane and lane+32)
  float qs0 = 0.f, qs1 = 0.f, ks0 = 0.f, ks1 = 0.f;
  for (int i = s; i < e; ++i) {
    const float* qr = Qn + (size_t)i * DD;
    const float* kr = Kn + (size_t)i * DD;
    qs0 += qr[lane];      qs1 += qr[lane + 32];
    ks0 += kr[lane];      ks1 += kr[lane + 32];
  }
  Qs[(size_t)p * DD + lane] = qs0;
  Qs[(size_t)p * DD + lane + 32] = qs1;
  const float inv = 1.0f / (float)((e - s) > 1 ? (e - s) : 1);
  Ksk[(size_t)p * DD + lane] = fmaxf(ks0 * inv, 0.0f);
  Ksk[(size_t)p * DD + lane + 32] = fmaxf(ks1 * inv, 0.0f);

  // M_p = relu(K)^T @ V : 4x4 grid of 16x16 f32 tiles, K stepped by 4 nodes
  const int m = lane & 15;
  const int kh = lane >> 4;
  const int n0 = lane & 15;
  v8f acc[16];
#pragma unroll
  for (int t = 0; t < 16; ++t)
#pragma unroll
    for (int r = 0; r < 8; ++r) acc[t][r] = 0.0f;

  for (int k0 = s; k0 < e; k0 += 4) {
    const int ka = k0 + 2 * kh;  // this lane's two K indices: ka, ka+1
    v2f a[4], b[4];
#pragma unroll
    for (int t = 0; t < 4; ++t) {
      const int af = t * 16 + m;   // M row (feature of K)
      a[t].x = (ka < e)     ? fmaxf(Kn[(size_t)ka * DD + af], 0.0f)       : 0.0f;
      a[t].y = (ka + 1 < e) ? fmaxf(Kn[(size_t)(ka + 1) * DD + af], 0.0f) : 0.0f;
      const int bf = t * 16 + n0;  // M col (feature of V)
      b[t].x = (ka < e)     ? Vn[(size_t)ka * DD + bf]       : 0.0f;
      b[t].y = (ka + 1 < e) ? Vn[(size_t)(ka + 1) * DD + bf] : 0.0f;
    }
#pragma unroll
    for (int ta = 0; ta < 4; ++ta)
#pragma unroll
      for (int tb = 0; tb < 4; ++tb)
        acc[ta * 4 + tb] = wmma4(a[ta], b[tb], acc[ta * 4 + tb]);
  }

  const int mhi = kh * 8;
  float* Mp = M + (size_t)p * (DD * DD);
#pragma unroll
  for (int ta = 0; ta < 4; ++ta)
#pragma unroll
    for (int tb = 0; tb < 4; ++tb)
#pragma unroll
      for (int r = 0; r < 8; ++r)
        Mp[(size_t)(ta * 16 + mhi + r) * DD + tb * 16 + n0] = acc[ta * 4 + tb][r];
}

// ---------------------------------------------------------------------------
// Fused edge-propagation + kernelized readout, one block (64 thr) per patch.
// Avoids materializing M2: num = Qk.M[p] + sum_e attr_e * (Qk.M[col_e]).
// ---------------------------------------------------------------------------
__global__ void patch_readout_kernel(const float* __restrict__ Qk,
                                     const float* __restrict__ Ksk,
                                     const float* __restrict__ M,
                                     const float* __restrict__ attrn,
                                     const int* __restrict__ rown,
                                     const int* __restrict__ coln,
                                     float* __restrict__ outP) {
  __shared__ int elist[MAXE];
  __shared__ float ewt[MAXE];
  __shared__ int ecnt;
  __shared__ float qk_s[DD];
  __shared__ float red[DD];
  const int p = blockIdx.x;
  const int t = threadIdx.x;  // 0..63

  if (t == 0) ecnt = 0;
  qk_s[t] = Qk[(size_t)p * DD + t];
  __syncthreads();
  for (int e = t; e < NEDGE; e += 64) {
    if (rown[e] == p) {
      int s = atomicAdd(&ecnt, 1);
      if (s < MAXE) { elist[s] = coln[e]; ewt[s] = attrn[e]; }
    }
  }
  __syncthreads();
  const int ne = ecnt < MAXE ? ecnt : MAXE;

  // Kagg feature t
  float kag = Ksk[(size_t)p * DD + t];
  for (int j = 0; j < ne; ++j) kag += ewt[j] * Ksk[(size_t)elist[j] * DD + t];

  // num[t] = Qk . M2[:, t]
  float num = 0.0f;
  {
    const float* Mp = M + (size_t)p * (DD * DD);
    for (int d = 0; d < DD; ++d) num += qk_s[d] * Mp[(size_t)d * DD + t];
  }
  for (int j = 0; j < ne; ++j) {
    const float* Mc = M + (size_t)elist[j] * (DD * DD);
    float part = 0.0f;
    for (int d = 0; d < DD; ++d) part += qk_s[d] * Mc[(size_t)d * DD + t];
    num += ewt[j] * part;
  }

  red[t] = qk_s[t] * kag;
  __syncthreads();
  float den = 0.0f;
  for (int d = 0; d < DD; ++d) den += red[d];
  outP[(size_t)p * DD + t] = num / (den + 1e-6f);
}

// ---------------------------------------------------------------------------
// Attention-pool over 32 patches per batch + 2-layer classifier. 1 wave / batch.
// ---------------------------------------------------------------------------
__global__ void final_kernel(const float* __restrict__ outP,
                             const float* __restrict__ seed,
                             const float* __restrict__ Wc1, const float* __restrict__ bc1,
                             const float* __restrict__ Wc2, const float* __restrict__ bc2,
                             float* __restrict__ logits) {
  __shared__ float scS[32], scE[32], scW[32];
  __shared__ float pooled[DD];
  __shared__ float hid[32];
  const int b = blockIdx.x, l = threadIdx.x;  // 32 threads
  const float* ob = outP + (size_t)b * 32 * DD;

  float s = 0.0f;
  for (int d = 0; d < DD; ++d) s += ob[(size_t)l * DD + d] * seed[d];
  s *= 0.125f;  // 1/sqrt(64)
  scS[l] = s;
  __syncthreads();
  float mx = -1e30f;
  for (int i = 0; i < 32; ++i) mx = fmaxf(mx, scS[i]);
  scE[l] = __expf(s - mx);
  __syncthreads();
  float sum = 0.0f;
  for (int i = 0; i < 32; ++i) sum += scE[i];
  scW[l] = scE[l] / sum;
  __syncthreads();

  float p0 = 0.0f, p1 = 0.0f;
  for (int p = 0; p < 32; ++p) {
    p0 += scW[p] * ob[(size_t)p * DD + l];
    p1 += scW[p] * ob[(size_t)p * DD + l + 32];
  }
  pooled[l] = p0;
  pooled[l + 32] = p1;
  __syncthreads();

  float hj = bc1[l];
  for (int d = 0; d < DD; ++d) hj += pooled[d] * Wc1[d * 32 + l];
  hid[l] = fmaxf(hj, 0.0f);
  __syncthreads();

  if (l < 10) {
    float lg = bc2[l];
    for (int j = 0; j < 32; ++j) lg += hid[j] * Wc2[j * 10 + l];
    logits[b * 10 + l] = lg;
  }
}

// ---------------------------------------------------------------------------
extern "C" void kernel_launch(void* const* d_in, const int* in_sizes, int n_in,
                              void* d_out, int out_size, void* d_ws, size_t ws_size,
                              hipStream_t stream) {
  (void)in_sizes; (void)n_in; (void)out_size; (void)ws_size;
  const float* x      = (const float*)d_in[0];
  const int*   mapper = (const int*)d_in[1];
  const int*   batch  = (const int*)d_in[2];
  const int*   rown   = (const int*)d_in[3];
  const int*   coln   = (const int*)d_in[4];
  const float* eattr  = (const float*)d_in[5];
  const float* Wt1 = (const float*)d_in[6];   const float* bt1 = (const float*)d_in[7];
  const float* Wt2 = (const float*)d_in[8];   const float* bt2 = (const float*)d_in[9];
  const float* Wpre1 = (const float*)d_in[10]; const float* bpre1 = (const float*)d_in[11];
  const float* Wpre2 = (const float*)d_in[12]; const float* bpre2 = (const float*)d_in[13];
  const float* Wpost1 = (const float*)d_in[14]; const float* bpost1 = (const float*)d_in[15];
  const float* Wpost2 = (const float*)d_in[16]; const float* bpost2 = (const float*)d_in[17];
  const float* Wk = (const float*)d_in[18];   const float* bk = (const float*)d_in[19];
  const float* Wv = (const float*)d_in[20];   const float* bv = (const float*)d_in[21];
  const float* seed = (const float*)d_in[22];
  const float* Wc1 = (const float*)d_in[23];  const float* bc1 = (const float*)d_in[24];
  const float* Wc2 = (const float*)d_in[25];  const float* bc2 = (const float*)d_in[26];

  char* ws = (char*)d_ws;
  size_t off = 0;
  auto alloc = [&](size_t bytes) -> void* {
    void* p = ws + off;
    off = (off + bytes + 255) & ~(size_t)255;
    return p;
  };
  float* h    = (float*)alloc((size_t)NNODES * DD * 4);
  float* Qn   = (float*)alloc((size_t)NNODES * DD * 4);
  float* Kn   = (float*)alloc((size_t)NNODES * DD * 4);
  float* Vn   = (float*)alloc((size_t)NNODES * DD * 4);
  float* M    = (float*)alloc((size_t)NPATCH_TOT * DD * DD * 4);
  float* Qs   = (float*)alloc((size_t)NPATCH_TOT * DD * 4);
  float* Qk   = (float*)alloc((size_t)NPATCH_TOT * DD * 4);
  float* Ksk  = (float*)alloc((size_t)NPATCH_TOT * DD * 4);
  float* outP = (float*)alloc((size_t)NPATCH_TOT * DD * 4);
  float* deg  = (float*)alloc((size_t)NPATCH_TOT * 4);
  float* attrn = (float*)alloc((size_t)NEDGE * 4);
  int* startp = (int*)alloc((size_t)(NPATCH_TOT + 1) * 4);

  // edge-weight symmetric normalization
  zero_kernel<<<(NPATCH_TOT + 255) / 256, 256, 0, stream>>>(deg, NPATCH_TOT);
  deg_accum_kernel<<<(NEDGE + 255) / 256, 256, 0, stream>>>(eattr, rown, deg);
  attr_norm_kernel<<<(NEDGE + 255) / 256, 256, 0, stream>>>(eattr, rown, coln, deg, attrn);
  patch_start_kernel<<<(NPATCH_TOT + 1 + 255) / 256, 256, 0, stream>>>(batch, startp);

  // node transform (WMMA): h = relu(relu(x@Wt1)@Wt2)
  mlp2rr_kernel<<<NNODES / 64, 128, 0, stream>>>(x, Wt1, bt1, Wt2, bt2, h);
  // gather + Q/K/V projections (WMMA)
  qkv_kernel<<<NNODES / 64, 128, 0, stream>>>(h, mapper, Wpre1, bpre1, Wpre2,
                                              bpre2, Wk, bk, Wv, bv, Qn, Kn, Vn);
  // per-patch sums + linear-attention state M (WMMA outer-product GEMM)
  patch_reduce_kernel<<<NPATCH_TOT, 32, 0, stream>>>(Qn, Kn, Vn, startp, M, Qs, Ksk);
  // post-deepset MLP + relu -> Qk (WMMA)
  mlp2rr_kernel<<<NPATCH_TOT / 64, 128, 0, stream>>>(Qs, Wpost1, bpost1, Wpost2,
                                                     bpost2, Qk);
  // edge propagation fused with kernelized attention readout
  patch_readout_kernel<<<NPATCH_TOT, 64, 0, stream>>>(Qk, Ksk, M, attrn, rown,
                                                      coln, outP);
  // attention pooling + classifier
  final_kernel<<<NBATCH, 32, 0, stream>>>(outP, seed, Wc1, bc1, Wc2, bc2,
                                          (float*)d_out);
}